// VolatilityEncodingModule_15109694948217
// MI455X (gfx1250) — compile-verified
//
#include <hip/hip_runtime.h>
#include <hip/hip_bf16.h>
#include <math.h>

typedef __attribute__((ext_vector_type(2))) float v2f;
typedef __attribute__((ext_vector_type(4))) float v4f;
typedef __attribute__((ext_vector_type(8))) float v8f;

#define S_LEN   4096      // sequence length (prices per row)
#define L_RET   (S_LEN-1) // returns per row = 4095
#define H2DIM   128
#define HID     256
#define LN_EPS  1e-5f

// ---------------------------------------------------------------------------
// Kernel 1: per-row volatility features + layer-1 (rank-3 matvec) + LN.
// One block (256 threads) per batch row. Memory bound: streams 16KB of
// prices per row (64 MiB total ~ 2.9us at 23.3 TB/s). Loads are b128 +
// non-temporal (read-once stream, keep L2 clean).
// ---------------------------------------------------------------------------
__global__ void __launch_bounds__(256)
vol_stage1_kernel(const float* __restrict__ prices,
                  const float* __restrict__ W1,   // [64,128]
                  const float* __restrict__ b1,
                  const float* __restrict__ g1,
                  const float* __restrict__ be1,
                  float* __restrict__ out)        // combined [B,256]
{
    __shared__ alignas(16) float lp[S_LEN];        // log prices
    __shared__ alignas(16) float ret[L_RET + 1];   // returns
    __shared__ float red0[256];
    __shared__ float red1[256];
    __shared__ float red2[256];
    __shared__ float hpre[H2DIM];
    __shared__ float volsh[3];
    __shared__ float stats[2];

    const int row = blockIdx.x;
    const int tid = threadIdx.x;
    const v4f* __restrict__ p4 =
        (const v4f*)(prices + (size_t)row * S_LEN);

    // log prices into LDS: 4 x global_load_b128 (NT) per thread, coalesced
    #pragma unroll
    for (int k = 0; k < S_LEN / (256 * 4); ++k) {
        int i4 = tid + k * 256;
        v4f v = __builtin_nontemporal_load(&p4[i4]);
        v4f l;
        l.x = __logf(fmaxf(v.x, 1e-8f));
        l.y = __logf(fmaxf(v.y, 1e-8f));
        l.z = __logf(fmaxf(v.z, 1e-8f));
        l.w = __logf(fmaxf(v.w, 1e-8f));
        *(v4f*)&lp[i4 * 4] = l;                    // ds_store_b128
    }
    __syncthreads();

    // returns
    #pragma unroll
    for (int k = 0; k < S_LEN / 256; ++k) {
        int i = tid + k * 256;
        if (i < L_RET) ret[i] = lp[i + 1] - lp[i];
    }
    __syncthreads();

    // sliding-window rolling std (ddof=1), one contiguous chunk per thread
    const int ws[3] = {5, 10, 20};
    float acc[3] = {0.f, 0.f, 0.f};
    #pragma unroll
    for (int wi = 0; wi < 3; ++wi) {
        const int w = ws[wi];
        const float invw = 1.0f / (float)w;
        const float invwm1 = 1.0f / (float)(w - 1);
        const int P = L_RET - w + 1;          // positions
        int p0 = tid * 16;
        int p1 = p0 + 16; if (p1 > P) p1 = P;
        if (p0 < p1) {
            float s1 = 0.f, s2 = 0.f;
            for (int i = p0; i < p0 + w; ++i) { float r = ret[i]; s1 += r; s2 += r * r; }
            int pos = p0;
            for (;;) {
                float var = (s2 - s1 * s1 * invw) * invwm1;
                acc[wi] += sqrtf(fmaxf(var, 0.f));
                ++pos; if (pos >= p1) break;
                float rn = ret[pos + w - 1];
                float ro = ret[pos - 1];
                s1 += rn - ro;
                s2 += rn * rn - ro * ro;
            }
        }
    }
    red0[tid] = acc[0]; red1[tid] = acc[1]; red2[tid] = acc[2];
    __syncthreads();
    for (int s = 128; s > 0; s >>= 1) {
        if (tid < s) { red0[tid] += red0[tid + s]; red1[tid] += red1[tid + s]; red2[tid] += red2[tid + s]; }
        __syncthreads();
    }
    if (tid < 3) {
        const int wv = (tid == 0) ? 5 : (tid == 1) ? 10 : 20;
        float sum = (tid == 0) ? red0[0] : (tid == 1) ? red1[0] : red2[0];
        volsh[tid] = sum / (float)(L_RET - wv + 1);
    }
    __syncthreads();

    // layer 1: vol(rank 3) @ W1 + b1 -> relu   (padded cols are zero)
    if (tid < H2DIM) {
        float x = b1[tid];
        x += volsh[0] * W1[0 * H2DIM + tid];
        x += volsh[1] * W1[1 * H2DIM + tid];
        x += volsh[2] * W1[2 * H2DIM + tid];
        hpre[tid] = fmaxf(x, 0.f);
    }
    __syncthreads();
    if (tid == 0) {
        float s1 = 0.f, s2 = 0.f;
        for (int j = 0; j < H2DIM; ++j) { float v = hpre[j]; s1 += v; s2 += v * v; }
        float mu = s1 * (1.0f / H2DIM);
        float var = s2 * (1.0f / H2DIM) - mu * mu;
        stats[0] = mu;
        stats[1] = rsqrtf(fmaxf(var, 0.f) + LN_EPS);
    }
    __syncthreads();
    if (tid < H2DIM) {
        float v = (hpre[tid] - stats[0]) * stats[1] * g1[tid] + be1[tid];
        out[(size_t)row * HID + tid] = v;   // first half of `combined`
    }
}

// ---------------------------------------------------------------------------
// Kernel 2: h @ W2 via V_WMMA_F32_16X16X4_F32 (+bias,relu,LN) -> h2f,
// then classifier head + softmax. One block (8 waves) per 16-row tile.
// Each wave owns one 16-wide N-tile of the 128 output columns.
// A-fragments come from a pitch-padded LDS tile (conflict-free ds_load_b64);
// B-fragments stream from global W2 (64 KB, L2-resident across 256 blocks).
// ---------------------------------------------------------------------------
#define APITCH 130   // LDS pitch padding to avoid 16-way bank conflicts

__global__ void __launch_bounds__(256)
mlp_stage2_kernel(float* __restrict__ out,        // d_out: combined + regime
                  const float* __restrict__ W2,   // [128,128]
                  const float* __restrict__ b2,
                  const float* __restrict__ g2,
                  const float* __restrict__ be2,
                  const float* __restrict__ Wc1,  // [256,64]
                  const float* __restrict__ bc1,
                  const float* __restrict__ Wc2,  // [64,4]
                  const float* __restrict__ bc2,
                  int B)
{
    __shared__ float At[16 * APITCH];   // h tile (first-half combined)
    __shared__ float H2[16 * APITCH];   // h2 pre-LN, then normalized h2f
    __shared__ float Hid[16 * 64];
    __shared__ float red[16 * 16 * 2];
    __shared__ float stats[16 * 2];
    __shared__ float Logit[16 * 4];

    const int tid  = threadIdx.x;
    const int row0 = blockIdx.x * 16;

    // warm W2 (64 KB) into cache: one prefetch per thread covers it all
    __builtin_prefetch(&W2[tid * 64], 0, 3);

    // stage h tile [16 x 128] from d_out (written by kernel 1)
    for (int idx = tid; idx < 16 * H2DIM; idx += 256) {
        int r = idx >> 7, c = idx & 127;
        At[r * APITCH + c] = out[(size_t)(row0 + r) * HID + c];
    }
    __syncthreads();

    // ---- GEMM: [16x128] x [128x128] with f32 WMMA 16x16x4 ----
    const int wave = tid >> 5;
    const int lane = tid & 31;
    const int half = lane >> 4;     // 0|1
    const int q    = lane & 15;     // M for A-frag, N for B/C-frag
    const int n0   = wave * 16;

    v8f acc = {};
    #pragma unroll 4
    for (int k0 = 0; k0 < H2DIM; k0 += 4) {
        const int ka = k0 + 2 * half;
        v2f a = *(const v2f*)&At[q * APITCH + ka];        // A[q][ka], A[q][ka+1]
        v2f b;
        b.x = W2[(size_t)ka * H2DIM + n0 + q];            // B[ka][n0+q]
        b.y = W2[(size_t)(ka + 1) * H2DIM + n0 + q];      // B[ka+1][n0+q]
        acc = __builtin_amdgcn_wmma_f32_16x16x4_f32(
            false, a, false, b, (short)0, acc, false, false);
    }
    // bias + relu, scatter C-frag to LDS: VGPR r -> row (r + 8*half), col n0+q
    {
        const int n = n0 + q;
        const float bias = b2[n];
        #pragma unroll
        for (int r = 0; r < 8; ++r) {
            int m = r + 8 * half;
            H2[m * APITCH + n] = fmaxf(acc[r] + bias, 0.f);
        }
    }
    __syncthreads();

    // ---- LayerNorm over 128 cols per row ----
    {
        const int r = tid >> 4, g16 = tid & 15;
        float s1 = 0.f, s2 = 0.f;
        #pragma unroll
        for (int i = 0; i < 8; ++i) {
            float v = H2[r * APITCH + g16 * 8 + i];
            s1 += v; s2 += v * v;
        }
        red[(r * 16 + g16) * 2 + 0] = s1;
        red[(r * 16 + g16) * 2 + 1] = s2;
    }
    __syncthreads();
    if (tid < 16) {
        float s1 = 0.f, s2 = 0.f;
        for (int g = 0; g < 16; ++g) { s1 += red[(tid * 16 + g) * 2]; s2 += red[(tid * 16 + g) * 2 + 1]; }
        float mu = s1 * (1.0f / H2DIM);
        float var = s2 * (1.0f / H2DIM) - mu * mu;
        stats[tid * 2 + 0] = mu;
        stats[tid * 2 + 1] = rsqrtf(fmaxf(var, 0.f) + LN_EPS);
    }
    __syncthreads();
    {
        const int r = tid >> 4, g16 = tid & 15;
        const float mu = stats[r * 2 + 0], rs = stats[r * 2 + 1];
        #pragma unroll
        for (int i = 0; i < 8; ++i) {
            int c = g16 * 8 + i;
            float v = (H2[r * APITCH + c] - mu) * rs * g2[c] + be2[c];
            H2[r * APITCH + c] = v;
            out[(size_t)(row0 + r) * HID + H2DIM + c] = v;  // second half of combined
        }
    }
    __syncthreads();

    // ---- classifier: relu(combined @ Wc1 + bc1) ----
    {
        const int r = tid >> 4, cg = tid & 15;
        float a0 = bc1[cg * 4 + 0], a1 = bc1[cg * 4 + 1];
        float a2 = bc1[cg * 4 + 2], a3 = bc1[cg * 4 + 3];
        for (int k = 0; k < HID; ++k) {
            float x = (k < H2DIM) ? At[r * APITCH + k] : H2[r * APITCH + (k - H2DIM)];
            const float* w = &Wc1[(size_t)k * 64 + cg * 4];
            a0 = fmaf(x, w[0], a0); a1 = fmaf(x, w[1], a1);
            a2 = fmaf(x, w[2], a2); a3 = fmaf(x, w[3], a3);
        }
        Hid[r * 64 + cg * 4 + 0] = fmaxf(a0, 0.f);
        Hid[r * 64 + cg * 4 + 1] = fmaxf(a1, 0.f);
        Hid[r * 64 + cg * 4 + 2] = fmaxf(a2, 0.f);
        Hid[r * 64 + cg * 4 + 3] = fmaxf(a3, 0.f);
    }
    __syncthreads();

    // ---- logits + softmax ----
    if (tid < 64) {
        const int r = tid >> 2, o = tid & 3;
        float s = bc2[o];
        for (int k = 0; k < 64; ++k) s = fmaf(Hid[r * 64 + k], Wc2[k * 4 + o], s);
        Logit[r * 4 + o] = s;
    }
    __syncthreads();
    if (tid < 16) {
        float l0 = Logit[tid * 4 + 0], l1 = Logit[tid * 4 + 1];
        float l2 = Logit[tid * 4 + 2], l3 = Logit[tid * 4 + 3];
        float m = fmaxf(fmaxf(l0, l1), fmaxf(l2, l3));
        float e0 = expf(l0 - m), e1 = expf(l1 - m), e2 = expf(l2 - m), e3 = expf(l3 - m);
        float inv = 1.0f / (e0 + e1 + e2 + e3);
        float* regime = out + (size_t)B * HID + (size_t)(row0 + tid) * 4;
        regime[0] = e0 * inv; regime[1] = e1 * inv;
        regime[2] = e2 * inv; regime[3] = e3 * inv;
    }
}

extern "C" void kernel_launch(void* const* d_in, const int* in_sizes, int n_in,
                              void* d_out, int out_size, void* d_ws, size_t ws_size,
                              hipStream_t stream) {
    const float* prices = (const float*)d_in[0];
    const float* W1  = (const float*)d_in[1];
    const float* b1  = (const float*)d_in[2];
    const float* g1  = (const float*)d_in[3];
    const float* be1 = (const float*)d_in[4];
    const float* W2  = (const float*)d_in[5];
    const float* b2  = (const float*)d_in[6];
    const float* g2  = (const float*)d_in[7];
    const float* be2 = (const float*)d_in[8];
    const float* Wc1 = (const float*)d_in[9];
    const float* bc1 = (const float*)d_in[10];
    const float* Wc2 = (const float*)d_in[11];
    const float* bc2 = (const float*)d_in[12];
    float* out = (float*)d_out;

    const int B = in_sizes[0] / S_LEN;   // 4096

    vol_stage1_kernel<<<B, 256, 0, stream>>>(prices, W1, b1, g1, be1, out);
    mlp_stage2_kernel<<<B / 16, 256, 0, stream>>>(out, W2, b2, g2, be2,
                                                  Wc1, bc1, Wc2, bc2, B);
}